// EqLayerESCNN_88656714925232
// MI455X (gfx1250) — compile-verified
//
#include <hip/hip_runtime.h>
#include <math.h>

// fp32 WMMA fragment types for V_WMMA_F32_16X16X4_F32:
// A (16x4 f32) = 2 VGPRs; B (4x16) = 2 VGPRs; C/D (16x16) = 8 VGPRs.
typedef float v2f __attribute__((ext_vector_type(2)));
typedef float v8f __attribute__((ext_vector_type(8)));

// LDS row strides (floats), padded for conflict-free b64 reads, 8/16B aligned.
#define SA 34   // 16 x 32 (a||b) tile
#define SS 68   // 16 x 64 scalar tile (16B-aligned rows for float4 stores)
#define SH 98   // 16 x 96 gated-hidden tile
#define SMEM_FLOATS (16 * SA + 16 * SS)   // 1632 floats; sH (1568) overlaps

// ---------------------------------------------------------------------------
// Folded complex-linear weight accessors.
//   W1c (32x96): rows 0..15 = [wr1 | wi1], rows 16..31 = [-wi1 | wr1]
//   W2c (96x32): rows 0..47 = [wr2 | wi2], rows 48..95 = [-wi2 | wr2]
// ---------------------------------------------------------------------------
__device__ __forceinline__ float foldW1(const float* wr1, const float* wi1, int k, int n) {
    if (k < 16) return (n < 48) ?  wr1[k * 48 + n]        : wi1[k * 48 + (n - 48)];
    else        return (n < 48) ? -wi1[(k - 16) * 48 + n] : wr1[(k - 16) * 48 + (n - 48)];
}
__device__ __forceinline__ float foldW2(const float* wr2, const float* wi2, int k, int n) {
    if (k < 48) return (n < 16) ?  wr2[k * 16 + n]        : wi2[k * 16 + (n - 16)];
    else        return (n < 16) ? -wi2[(k - 48) * 16 + n] : wr2[(k - 48) * 16 + (n - 16)];
}

// ---------------------------------------------------------------------------
// Prep: emit B matrices pre-swizzled into per-lane WMMA fragment order.
// frag[(n*K4 + k)*32 + lane] = { B[4k + 2*(lane>>4)][16n + (lane&15)],
//                                B[4k + 2*(lane>>4) + 1][...] }
// Main kernel then fetches each B fragment with ONE coalesced global_load_b64.
// ---------------------------------------------------------------------------
__global__ void prep_frags(const float* __restrict__ wr1, const float* __restrict__ wi1,
                           const float* __restrict__ wr2, const float* __restrict__ wi2,
                           const float* __restrict__ wsg,
                           float* __restrict__ w1f,   // 6*8*32  v2f
                           float* __restrict__ wgf,   // 3*16*32 v2f
                           float* __restrict__ w2f) { // 2*24*32 v2f
    int i = blockIdx.x * blockDim.x + threadIdx.x;
    if (i >= 3 * 1536) return;
    int region = i / 1536;
    int f      = i - region * 1536;
    int lane   = f & 31;
    int step   = f >> 5;
    int kb     = 2 * (lane >> 4);
    int nc     = lane & 15;
    if (region == 0) {                       // GEMM1: 6 n-tiles x 8 k-steps
        int n = step >> 3, k = step & 7;
        int kk = 4 * k + kb, nn = n * 16 + nc;
        w1f[2 * f]     = foldW1(wr1, wi1, kk,     nn);
        w1f[2 * f + 1] = foldW1(wr1, wi1, kk + 1, nn);
    } else if (region == 1) {                // gate: 3 n-tiles x 16 k-steps
        int n = step >> 4, k = step & 15;
        int kk = 4 * k + kb, nn = n * 16 + nc;
        wgf[2 * f]     = wsg[kk * 48 + nn];
        wgf[2 * f + 1] = wsg[(kk + 1) * 48 + nn];
    } else {                                 // GEMM2: 2 n-tiles x 24 k-steps
        int n = step / 24, k = step % 24;
        int kk = 4 * k + kb, nn = n * 16 + nc;
        w2f[2 * f]     = foldW2(wr2, wi2, kk,     nn);
        w2f[2 * f + 1] = foldW2(wr2, wi2, kk + 1, nn);
    }
}

__global__ void zero_kernel(float* __restrict__ p, int n) {
    int i = blockIdx.x * blockDim.x + threadIdx.x;
    if (i < n) p[i] = 0.0f;
}

// ---------------------------------------------------------------------------
// Main: one wave32 per 16-edge tile. Three chained fp32 WMMA GEMMs:
//   (16x32)x(32x96) -> gate(16x64)x(64x48) silu -> (16x96)x(96x32) -> scatter.
// ---------------------------------------------------------------------------
__global__ __launch_bounds__(32)
void escnn_edge_kernel(const float* __restrict__ x_scalar,
                       const float* __restrict__ x_rot,
                       const int*   __restrict__ edge_row,
                       const int*   __restrict__ edge_col,
                       const float* __restrict__ dist_emb,
                       const float* __restrict__ bs,    // (48)
                       const v2f*   __restrict__ w1f,
                       const v2f*   __restrict__ wgf,
                       const v2f*   __restrict__ w2f,
                       float* __restrict__ mess,        // (N x 32) accum
                       int E)
{
    __shared__ float smem[SMEM_FLOATS];
    __shared__ int   sCol[16];
    float* sA = smem;              // 16*SA
    float* sS = smem + 16 * SA;    // 16*SS (16B-aligned: 544*4 bytes)
    float* sH = smem;              // reused once aA/aS are in registers

    const int lane = threadIdx.x;
    const int e0   = blockIdx.x * 16;
    if (e0 >= E) return;                // uniform per wave: EXEC stays all-1s for WMMA

    const int h    = lane >> 4;         // lane half (0/1) -> K sub-offset in fragments
    const int el   = lane & 15;         // local edge / N / M index
    const int ge   = e0 + el;
    const bool valid = (ge < E);
    const int r    = valid ? edge_row[ge] : 0;
    if (h == 0) sCol[el] = valid ? edge_col[ge] : -1;

    // ---- Stage gathered inputs into LDS --------------------------------
    {   // x_rot[r] (interleaved a0,b0,...) -> sA row el as (a0..a15, b0..b15)
        const float4* src = (const float4*)(x_rot + (size_t)r * 32 + h * 16);
        #pragma unroll
        for (int q = 0; q < 4; ++q) {
            float4 v = valid ? src[q] : make_float4(0.f, 0.f, 0.f, 0.f);
            int p = h * 8 + 2 * q;
            sA[el * SA + p]      = v.x;
            sA[el * SA + 16 + p] = v.y;
            sA[el * SA + p + 1]  = v.z;
            sA[el * SA + 17 + p] = v.w;
        }
    }
    {   // scal = [dist_emb[e] | x_scalar[r]] -> sS row el
        const float4* d = (const float4*)(dist_emb + (size_t)ge * 32 + h * 16);
        const float4* s = (const float4*)(x_scalar + (size_t)r  * 32 + h * 16);
        #pragma unroll
        for (int q = 0; q < 4; ++q) {
            float4 dv = valid ? d[q] : make_float4(0.f, 0.f, 0.f, 0.f);
            float4 sv = valid ? s[q] : make_float4(0.f, 0.f, 0.f, 0.f);
            *(float4*)&sS[el * SS +      h * 16 + 4 * q] = dv;
            *(float4*)&sS[el * SS + 32 + h * 16 + 4 * q] = sv;
        }
    }
    __syncthreads();

    const int ma   = el;       // A-fragment row owned by this lane
    const int koff = 2 * h;    // lane-half K offset (ISA A/B layout for 16x16x4)

    // ---- Preload ALL A-fragments (smem gets recycled for sH) -----------
    v2f aA[8];
    #pragma unroll
    for (int k = 0; k < 8; ++k)
        aA[k] = *(const v2f*)&sA[ma * SA + 4 * k + koff];
    v2f aS[16];
    #pragma unroll
    for (int k = 0; k < 16; ++k)
        aS[k] = *(const v2f*)&sS[ma * SS + 4 * k + koff];
    __syncthreads();

    // ---- GEMM1: (16x32) x (32x96) -> hc[0..5] (hr | hi) ----------------
    v8f hc[6];
    #pragma unroll
    for (int n = 0; n < 6; ++n) {
        v8f c = {};
        #pragma unroll
        for (int k = 0; k < 8; ++k) {
            v2f b = w1f[(n * 8 + k) * 32 + lane];   // one coalesced b64 load
            c = __builtin_amdgcn_wmma_f32_16x16x4_f32(false, aA[k], false, b,
                                                      (short)0, c, false, false);
        }
        hc[n] = c;
    }

    // ---- Gate GEMM: (16x64) x (64x48) -> cg[0..2] ----------------------
    v8f cg[3];
    #pragma unroll
    for (int n = 0; n < 3; ++n) {
        v8f c = {};
        #pragma unroll
        for (int k = 0; k < 16; ++k) {
            v2f b = wgf[(n * 16 + k) * 32 + lane];
            c = __builtin_amdgcn_wmma_f32_16x16x4_f32(false, aS[k], false, b,
                                                      (short)0, c, false, false);
        }
        cg[n] = c;
    }

    // ---- bias + silu, gate hr/hi in C-fragment layout ------------------
    #pragma unroll
    for (int n = 0; n < 3; ++n) {
        float bb = bs[n * 16 + el];            // C layout: column = lane&15
        #pragma unroll
        for (int j = 0; j < 8; ++j) {
            float z = cg[n][j] + bb;
            float g = z / (1.0f + __expf(-z)); // silu
            hc[n][j]     *= g;
            hc[n + 3][j] *= g;
        }
    }

    // ---- Re-fragment gated hidden via LDS (overlaps dead sA/sS) --------
    // C layout: VGPR j holds row M=j (lanes 0-15) / M=j+8 (lanes 16-31), col = el
    #pragma unroll
    for (int t = 0; t < 6; ++t)
        #pragma unroll
        for (int j = 0; j < 8; ++j)
            sH[(j + 8 * h) * SH + t * 16 + el] = hc[t][j];
    __syncthreads();

    // ---- GEMM2: (16x96) x (96x32) -> c2[0..1] (orr | oii) --------------
    v8f c2[2];
    #pragma unroll
    for (int n = 0; n < 2; ++n) {
        v8f c = {};
        #pragma unroll
        for (int k = 0; k < 24; ++k) {
            int kb = 4 * k + koff;
            v2f a = *(const v2f*)&sH[ma * SH + kb];
            v2f b = w2f[(n * 24 + k) * 32 + lane];
            c = __builtin_amdgcn_wmma_f32_16x16x4_f32(false, a, false, b,
                                                      (short)0, c, false, false);
        }
        c2[n] = c;
    }

    // ---- Scatter-add to destination nodes ------------------------------
    // x_out[e][2r + t]: t=0 -> orr (tile 0), t=1 -> oii (tile 1), r = el
    #pragma unroll
    for (int t = 0; t < 2; ++t)
        #pragma unroll
        for (int j = 0; j < 8; ++j) {
            int m = j + 8 * h;
            int cidx = sCol[m];
            if (cidx >= 0)
                atomicAdd(&mess[(size_t)cidx * 32 + el * 2 + t], c2[t][j]);
        }
}

// ---------------------------------------------------------------------------
extern "C" void kernel_launch(void* const* d_in, const int* in_sizes, int n_in,
                              void* d_out, int out_size, void* d_ws, size_t ws_size,
                              hipStream_t stream) {
    const float* x_scalar = (const float*)d_in[0];
    const float* x_rot    = (const float*)d_in[1];
    const int*   eidx     = (const int*)  d_in[2];
    const float* de       = (const float*)d_in[3];
    // d_in[4] = rot, unused by the reference math
    const float* wr1 = (const float*)d_in[5];
    const float* wi1 = (const float*)d_in[6];
    const float* wr2 = (const float*)d_in[7];
    const float* wi2 = (const float*)d_in[8];
    const float* wsg = (const float*)d_in[9];
    const float* bs  = (const float*)d_in[10];

    const int Nn = in_sizes[0] / 32;   // nodes
    const int E  = in_sizes[2] / 2;    // edges

    float* w1f = (float*)d_ws;         // 6*8*32  v2f = 3072 floats
    float* wgf = w1f + 3072;           // 3*16*32 v2f = 3072 floats
    float* w2f = wgf + 3072;           // 2*24*32 v2f = 3072 floats

    float* out_scalar = (float*)d_out;
    float* mess = out_scalar + (size_t)Nn * 32;

    // d_out = [x_scalar passthrough | zero-initialized message accumulator]
    (void)hipMemcpyAsync(out_scalar, x_scalar, (size_t)Nn * 32 * sizeof(float),
                         hipMemcpyDeviceToDevice, stream);
    zero_kernel<<<(Nn * 32 + 255) / 256, 256, 0, stream>>>(mess, Nn * 32);

    prep_frags<<<(3 * 1536 + 255) / 256, 256, 0, stream>>>(wr1, wi1, wr2, wi2, wsg,
                                                           w1f, wgf, w2f);

    int tiles = (E + 15) / 16;
    escnn_edge_kernel<<<tiles, 32, 0, stream>>>(x_scalar, x_rot, eidx, eidx + E,
                                                de, bs,
                                                (const v2f*)w1f, (const v2f*)wgf,
                                                (const v2f*)w2f, mess, E);
}